// SelfTp_24644522344488
// MI455X (gfx1250) — compile-verified
//
#include <hip/hip_runtime.h>

typedef __attribute__((ext_vector_type(2))) float v2f;
typedef __attribute__((ext_vector_type(8))) float v8f;

#define WAVES_PER_BLOCK 4
#define ROW_STRIDE 244  // padded LDS row stride (floats): 16B aligned, bank-conflict free

struct TpWeights { const float* w1[8]; const float* w2[8]; };

static __device__ __forceinline__ v8f vzero8() {
  v8f z = {0.f, 0.f, 0.f, 0.f, 0.f, 0.f, 0.f, 0.f};
  return z;
}

// t[w, comp i] = sum_u x[u, i] * w[u, w] for a 16-row z-tile, one 16-wide w tile.
// A tile: 16 z-rows x 4 K (from LDS); B tile: 4 K x 16 w (weights, L2-resident).
// acc[i] holds t[z, w, i] in WMMA C/D layout: lane L, reg r <-> z = r + 8*(L>>4), w = L&15.
template <int NCOMP, int MIN>
static __device__ __forceinline__ void tp_gemm(const float* __restrict__ xrow,
                                               const float* __restrict__ w,
                                               const int MO, const int wc,
                                               const int kh2, v8f* acc) {
#pragma unroll
  for (int i = 0; i < NCOMP; ++i) acc[i] = vzero8();
#pragma unroll
  for (int kc = 0; kc < MIN / 4; ++kc) {
    const int k0 = 4 * kc + kh2;  // A: vgpr0 holds K = {0,2}, vgpr1 K = {1,3} per half-wave
    v2f b;
    b.x = w[(k0    ) * MO + wc];
    b.y = w[(k0 + 1) * MO + wc];
#pragma unroll
    for (int i = 0; i < NCOMP; ++i) {  // reuse B across spherical components
      v2f a;
      a.x = xrow[(k0    ) * NCOMP + i];
      a.y = xrow[(k0 + 1) * NCOMP + i];
      acc[i] = __builtin_amdgcn_wmma_f32_16x16x4_f32(false, a, false, b,
                                                     (short)0, acc[i], false, false);
    }
  }
}

__global__ __launch_bounds__(WAVES_PER_BLOCK * 32) void selftp_wmma_kernel(
    const float* __restrict__ x, TpWeights W, float* __restrict__ out) {
  __shared__ float xs_all[WAVES_PER_BLOCK * 16 * ROW_STRIDE];

  const int lane = threadIdx.x & 31;
  const int wave = threadIdx.x >> 5;
  const int z0 = ((int)blockIdx.x * WAVES_PER_BLOCK + wave) * 16;
  float* xs = xs_all + wave * (16 * ROW_STRIDE);

  // Stage this wave's 16x240 x-tile into LDS (float4, padded stride).
  for (int t = lane; t < 16 * 60; t += 32) {
    const int r = t / 60, c = t % 60;
    const float4 v = *(const float4*)(x + (size_t)(z0 + r) * 240 + c * 4);
    *(float4*)(xs + r * ROW_STRIDE + c * 4) = v;
  }
  __syncthreads();

  const int m = lane & 15;        // z-row within half, also w-column index
  const int kh = lane >> 4;       // half-wave select
  const int kh2 = kh << 1;
  const float* xr0 = xs + m * ROW_STRIDE;        // 64x l=0  (cols 0..63)
  const float* xr1 = xr0 + 64;                   // 32x l=1  (cols 64..159)
  const float* xr2 = xr0 + 160;                  // 16x l=2  (cols 160..239)

  v8f a1[5], a2[5], o[5];

  // w3j constants (real basis, Frobenius-norm 1) and path normalizations.
  constexpr float C110 = 0.57735026918962576f;   // 1/sqrt(3)   (1,1,0) & (0,1,1)
  constexpr float C220 = 0.44721359549995794f;   // 1/sqrt(5)   (2,2,0) & (0,2,2)
  constexpr float C10  = 0.31622776601683794f;   // 1/sqrt(10)
  constexpr float C30  = 0.18257418583505536f;   // 1/sqrt(30)
  constexpr float C30x2= 0.36514837167011072f;   // 2/sqrt(30)
  constexpr float N1   = 0.23904572186687872f;   // sqrt(2/35)
  constexpr float N2   = 0.20701966780270626f;   // sqrt(3/70)
  constexpr float N3   = 0.11952286093343936f;   // 1/sqrt(70)
  constexpr float AL0  = 0.013638620234691170f;  // sqrt(1/5376)
  constexpr float AL1  = 0.034232659844072866f;  // sqrt(3/2560)
  constexpr float AL2  = 0.046585084738934310f;  // sqrt(5/2304)

  // ================= io = 0 : 64x l=0 (out cols 0..63) =================
  for (int wt = 0; wt < 4; ++wt) {
    const int wc = wt * 16 + m;
    o[0] = vzero8();
    // path (0,0,0)
    tp_gemm<1, 64>(xr0, W.w1[0], 64, wc, kh2, a1);
    tp_gemm<1, 64>(xr0, W.w2[0], 64, wc, kh2, a2);
    o[0] += a1[0] * a2[0];
    // path (1,1,0)
    tp_gemm<3, 32>(xr1, W.w1[3], 64, wc, kh2, a1);
    tp_gemm<3, 32>(xr1, W.w2[3], 64, wc, kh2, a2);
    o[0] += C110 * (a1[0] * a2[0] + a1[1] * a2[1] + a1[2] * a2[2]);
    // path (2,2,0)
    tp_gemm<5, 16>(xr2, W.w1[6], 64, wc, kh2, a1);
    tp_gemm<5, 16>(xr2, W.w2[6], 64, wc, kh2, a2);
    o[0] += C220 * (a1[0] * a2[0] + a1[1] * a2[1] + a1[2] * a2[2] +
                    a1[3] * a2[3] + a1[4] * a2[4]);
#pragma unroll
    for (int r = 0; r < 8; ++r)
      out[(size_t)(z0 + r + 8 * kh) * 240 + wc] = AL0 * o[0][r];
  }

  // ================= io = 1 : 32x l=1 (out cols 64..159) =================
  for (int wt = 0; wt < 2; ++wt) {
    const int wc = wt * 16 + m;
    o[0] = o[1] = o[2] = vzero8();
    // path (0,1,1)
    tp_gemm<1, 64>(xr0, W.w1[1], 32, wc, kh2, a1);
    tp_gemm<3, 32>(xr1, W.w2[1], 32, wc, kh2, a2);
    o[0] += C110 * (a1[0] * a2[0]);
    o[1] += C110 * (a1[0] * a2[1]);
    o[2] += C110 * (a1[0] * a2[2]);
    // path (1,2,1)
    tp_gemm<3, 32>(xr1, W.w1[5], 32, wc, kh2, a1);
    tp_gemm<5, 16>(xr2, W.w2[5], 32, wc, kh2, a2);
    o[0] += C10 * (a1[2] * a2[0] + a1[1] * a2[1] - a1[0] * a2[4]) - C30 * (a1[0] * a2[2]);
    o[1] += C10 * (a1[0] * a2[1] + a1[2] * a2[3]) + C30x2 * (a1[1] * a2[2]);
    o[2] += C10 * (a1[0] * a2[0] + a1[1] * a2[3] + a1[2] * a2[4]) - C30 * (a1[2] * a2[2]);
#pragma unroll
    for (int r = 0; r < 8; ++r) {
      float* p = out + (size_t)(z0 + r + 8 * kh) * 240 + 64 + wc * 3;
      p[0] = AL1 * o[0][r];
      p[1] = AL1 * o[1][r];
      p[2] = AL1 * o[2][r];
    }
  }

  // ================= io = 2 : 16x l=2 (out cols 160..239) =================
  {
    const int wc = m;
    o[0] = o[1] = o[2] = o[3] = o[4] = vzero8();
    // path (0,2,2)
    tp_gemm<1, 64>(xr0, W.w1[2], 16, wc, kh2, a1);
    tp_gemm<5, 16>(xr2, W.w2[2], 16, wc, kh2, a2);
#pragma unroll
    for (int k = 0; k < 5; ++k) o[k] += C220 * (a1[0] * a2[k]);
    // path (1,1,2)
    tp_gemm<3, 32>(xr1, W.w1[4], 16, wc, kh2, a1);
    tp_gemm<3, 32>(xr1, W.w2[4], 16, wc, kh2, a2);
    o[0] += C10 * (a1[2] * a2[0] + a1[0] * a2[2]);
    o[1] += C10 * (a1[0] * a2[1] + a1[1] * a2[0]);
    o[2] += C30x2 * (a1[1] * a2[1]) - C30 * (a1[0] * a2[0] + a1[2] * a2[2]);
    o[3] += C10 * (a1[2] * a2[1] + a1[1] * a2[2]);
    o[4] += C10 * (a1[2] * a2[2] - a1[0] * a2[0]);
    // path (2,2,2)
    tp_gemm<5, 16>(xr2, W.w1[7], 16, wc, kh2, a1);
    tp_gemm<5, 16>(xr2, W.w2[7], 16, wc, kh2, a2);
    o[0] += -N1 * (a1[0] * a2[2] + a1[2] * a2[0]) + N2 * (a1[1] * a2[3] + a1[3] * a2[1]);
    o[1] +=  N2 * (a1[0] * a2[3] + a1[3] * a2[0] - a1[1] * a2[4] - a1[4] * a2[1]) +
             N3 * (a1[1] * a2[2] + a1[2] * a2[1]);
    o[2] +=  N3 * (a1[1] * a2[1] + a1[3] * a2[3]) +
             N1 * (a1[2] * a2[2] - a1[0] * a2[0] - a1[4] * a2[4]);
    o[3] +=  N2 * (a1[0] * a2[1] + a1[1] * a2[0] + a1[3] * a2[4] + a1[4] * a2[3]) +
             N3 * (a1[2] * a2[3] + a1[3] * a2[2]);
    o[4] += -N1 * (a1[2] * a2[4] + a1[4] * a2[2]) + N2 * (a1[3] * a2[3] - a1[1] * a2[1]);
#pragma unroll
    for (int r = 0; r < 8; ++r) {
      float* p = out + (size_t)(z0 + r + 8 * kh) * 240 + 160 + wc * 5;
#pragma unroll
      for (int k = 0; k < 5; ++k) p[k] = AL2 * o[k][r];
    }
  }
}

extern "C" void kernel_launch(void* const* d_in, const int* in_sizes, int n_in,
                              void* d_out, int out_size, void* d_ws, size_t ws_size,
                              hipStream_t stream) {
  (void)n_in; (void)out_size; (void)d_ws; (void)ws_size;
  const float* x = (const float*)d_in[0];
  TpWeights W;
  for (int p = 0; p < 8; ++p) {
    W.w1[p] = (const float*)d_in[1 + 2 * p];
    W.w2[p] = (const float*)d_in[2 + 2 * p];
  }
  float* out = (float*)d_out;
  const int n = in_sizes[0] / 240;                  // 262144
  const int rows_per_block = WAVES_PER_BLOCK * 16;  // 64
  dim3 grid(n / rows_per_block), block(WAVES_PER_BLOCK * 32);
  hipLaunchKernelGGL(selftp_wmma_kernel, grid, block, 0, stream, x, W, out);
}